// TensorDLT_154618822721
// MI455X (gfx1250) — compile-verified
//
#include <hip/hip_runtime.h>
#include <stdint.h>

// ---------------------------------------------------------------------------
// TensorDLT: per-element 8x8 DLT solve, h = A^{-1} b, H = [h;1] (3x3).
// B = 524288 independent systems.
//  - Inputs staged via Tensor Data Mover (tensor_load_to_lds, padded rows).
//  - Register-resident unrolled Gauss-Jordan with cndmask partial pivoting.
//  - Outputs staged in LDS and written back as one contiguous TDM DMA per
//    block (tensor_store_from_lds), drained with s_wait_tensorcnt.
// ---------------------------------------------------------------------------

typedef __attribute__((ext_vector_type(4))) unsigned int u32x4;
typedef __attribute__((ext_vector_type(4))) int          i32x4;
typedef __attribute__((ext_vector_type(8))) int          i32x8;

#define BLK      256
#define ROW_PAD  10   // 8 payload floats + 2 pad floats (TDM pad: 2 DW per 8 DW)
#define OUT_ROW  9    // contiguous output rows (no de-padding on TDM store)

#if __has_builtin(__builtin_amdgcn_tensor_load_to_lds) && \
    __has_builtin(__builtin_amdgcn_s_wait_tensorcnt)
#define USE_TDM_LD 1
#else
#define USE_TDM_LD 0
#endif

#if __has_builtin(__builtin_amdgcn_tensor_store_from_lds) && \
    __has_builtin(__builtin_amdgcn_s_wait_tensorcnt)
#define USE_TDM_ST 1
#else
#define USE_TDM_ST 0
#endif

#if USE_TDM_LD || USE_TDM_ST
// Build a 1-D TDM descriptor: n_elems f32 at LDS byte offset lds_off <->
// global address ga. Optional LDS padding: +2 DWORDs after every 8 DWORDs.
__device__ __forceinline__ void tdm_desc_1d(u32x4& g0, i32x8& g1,
                                            uint32_t lds_off, uint64_t ga,
                                            uint32_t n_elems, bool pad)
{
    g0[0] = 1u;                                        // count=1 (valid), user mode
    g0[1] = lds_off;                                   // lds_addr (bytes)
    g0[2] = (uint32_t)(ga & 0xFFFFFFFFull);            // global_addr[31:0]
    g0[3] = (uint32_t)((ga >> 32) & 0x1FFFFFFull)      // global_addr[56:32]
          | (2u << 30);                                // type = 2 ("image")

    uint32_t w0 = (2u << 16);                          // data_size = 4 bytes
    if (pad) w0 |= (1u << 20)                          // pad_enable
                 | (2u << 22)                          // pad_interval: every 8 DWORDs
                 | (1u << 25);                         // pad_amount: 2 DWORDs
    g1[0] = (int)w0;
    g1[1] = (int)((n_elems & 0xFFFFu) << 16);          // tensor_dim0[15:0]
    g1[2] = (int)(n_elems >> 16);                      // tensor_dim0[31:16]; tensor_dim1=0
    g1[3] = (int)((n_elems & 0xFFFFu) << 16);          // tile_dim0 = n_elems (<= 2304)
    g1[4] = 0;                                         // tile_dim1 = tile_dim2 = 0 (1-D)
    g1[5] = (int)n_elems;                              // tensor_dim0_stride[31:0]
    g1[6] = 0;
    g1[7] = 0;
}
#endif

#if USE_TDM_LD
__device__ __forceinline__ void tdm_load_1d(uint32_t lds_off, const float* gptr,
                                            uint32_t n_elems)
{
    u32x4 g0; i32x8 g1;
    tdm_desc_1d(g0, g1, lds_off, (uint64_t)(uintptr_t)gptr, n_elems, /*pad=*/true);
    i32x4 z4 = (i32x4)0;
#if defined(__clang_major__) && (__clang_major__ >= 23)
    i32x8 z8 = (i32x8)0;
    __builtin_amdgcn_tensor_load_to_lds(g0, g1, z4, z4, z8, 0);
#else
    __builtin_amdgcn_tensor_load_to_lds(g0, g1, z4, z4, 0);
#endif
}
#endif

#if USE_TDM_ST
__device__ __forceinline__ void tdm_store_1d(uint32_t lds_off, float* gptr,
                                             uint32_t n_elems)
{
    u32x4 g0; i32x8 g1;
    tdm_desc_1d(g0, g1, lds_off, (uint64_t)(uintptr_t)gptr, n_elems, /*pad=*/false);
    i32x4 z4 = (i32x4)0;
#if defined(__clang_major__) && (__clang_major__ >= 23)
    i32x8 z8 = (i32x8)0;
    __builtin_amdgcn_tensor_store_from_lds(g0, g1, z4, z4, z8, 0);
#else
    __builtin_amdgcn_tensor_store_from_lds(g0, g1, z4, z4, 0);
#endif
}
#endif

__global__ __launch_bounds__(BLK)
void tensor_dlt_kernel(const float* __restrict__ ca,
                       const float* __restrict__ h4,
                       float* __restrict__ out,
                       int B)
{
    __shared__ float lca[BLK * ROW_PAD];   // 10240 B; reused for output staging
    __shared__ float lhp[BLK * ROW_PAD];   // 10240 B

    const int t    = (int)threadIdx.x;
    const int base = (int)blockIdx.x * BLK;

    const int rem = B - base;
    const uint32_t n_this  = (uint32_t)(rem < BLK ? rem : BLK);
    const uint32_t n_elems = n_this * 8u;

    // ---------------- input staging ----------------
#if USE_TDM_LD
    if ((t >> 5) == 0) {   // wave 0 issues both DMAs for the whole block
        tdm_load_1d((uint32_t)(uintptr_t)&lca[0], ca + (size_t)base * 8, n_elems);
        tdm_load_1d((uint32_t)(uintptr_t)&lhp[0], h4 + (size_t)base * 8, n_elems);
        __builtin_amdgcn_s_wait_tensorcnt(0);
    }
    __syncthreads();
#else
    for (uint32_t e = (uint32_t)t; e < n_elems; e += BLK) {
        lca[(e >> 3) * ROW_PAD + (e & 7u)] = ca[(size_t)base * 8 + e];
        lhp[(e >> 3) * ROW_PAD + (e & 7u)] = h4[(size_t)base * 8 + e];
    }
    __syncthreads();
#endif

    // ---------------- load my 16 inputs into registers ----------------
    const float* cv = &lca[t * ROW_PAD];   // stride-10 rows: bank-conflict-free
    const float* hv = &lhp[t * ROW_PAD];

    float x[4], y[4], xp[4], yp[4];
#pragma unroll
    for (int i = 0; i < 4; ++i) {
        x[i]  = cv[2 * i];  y[i]  = cv[2 * i + 1];
        xp[i] = hv[2 * i];  yp[i] = hv[2 * i + 1];
    }

#if USE_TDM_ST
    // Inputs are now in VGPRs (barrier fences DScnt); lca becomes output stage.
    __syncthreads();
#endif

    // ---------------- build augmented [A | b] in VGPRs ----------------
    float M[8][9];
#pragma unroll
    for (int i = 0; i < 4; ++i) {
        float* r1 = M[2 * i];       // row1_i, rhs = xp_i
        float* r2 = M[2 * i + 1];   // row2_i, rhs = yp_i
        r1[0] = 0.f;        r1[1] = 0.f;        r1[2] = 0.f;
        r1[3] = -x[i];      r1[4] = -y[i];      r1[5] = -1.f;
        r1[6] = yp[i] * x[i];   r1[7] = yp[i] * y[i];   r1[8] = xp[i];
        r2[0] = x[i];       r2[1] = y[i];       r2[2] = 1.f;
        r2[3] = 0.f;        r2[4] = 0.f;        r2[5] = 0.f;
        r2[6] = -xp[i] * x[i];  r2[7] = -xp[i] * y[i];  r2[8] = yp[i];
    }

    // ---------------- Gauss-Jordan, cndmask partial pivoting ----------------
#pragma unroll
    for (int k = 0; k < 8; ++k) {
#pragma unroll
        for (int j = k + 1; j < 8; ++j) {
            const bool sw = __builtin_fabsf(M[j][k]) > __builtin_fabsf(M[k][k]);
#pragma unroll
            for (int c = k; c < 9; ++c) {
                const float a = M[k][c], b = M[j][c];
                M[k][c] = sw ? b : a;
                M[j][c] = sw ? a : b;
            }
        }
        const float inv = 1.0f / M[k][k];
#pragma unroll
        for (int c = k + 1; c < 9; ++c) M[k][c] *= inv;
        M[k][k] = 1.0f;
#pragma unroll
        for (int j = 0; j < 8; ++j) {
            if (j == k) continue;
            const float f = M[j][k];
            M[j][k] = 0.0f;
#pragma unroll
            for (int c = k + 1; c < 9; ++c)
                M[j][c] = __builtin_fmaf(-f, M[k][c], M[j][c]);
        }
    }

    // ---------------- output ----------------
#if USE_TDM_ST
    // Contiguous rows of 9 (odd stride -> conflict-free ds_store), then one
    // TDM DMA of the whole block's 256*9 floats.
    if (t < (int)n_this) {
        float* ov = &lca[t * OUT_ROW];
#pragma unroll
        for (int i = 0; i < 8; ++i) ov[i] = M[i][8];
        ov[8] = 1.0f;
    }
    __syncthreads();
    if ((t >> 5) == 0) {
        tdm_store_1d((uint32_t)(uintptr_t)&lca[0],
                     out + (size_t)base * OUT_ROW, n_this * (uint32_t)OUT_ROW);
        __builtin_amdgcn_s_wait_tensorcnt(0);
    }
#else
    if (t < (int)n_this) {
        float* o = out + (size_t)(base + t) * OUT_ROW;
#pragma unroll
        for (int i = 0; i < 8; ++i) o[i] = M[i][8];
        o[8] = 1.0f;
    }
#endif
}

extern "C" void kernel_launch(void* const* d_in, const int* in_sizes, int n_in,
                              void* d_out, int out_size, void* d_ws, size_t ws_size,
                              hipStream_t stream)
{
    (void)n_in; (void)out_size; (void)d_ws; (void)ws_size;
    const float* ca = (const float*)d_in[0];
    const float* h4 = (const float*)d_in[1];
    float* out = (float*)d_out;

    const int B = in_sizes[0] / 8;          // 524288
    const int blocks = (B + BLK - 1) / BLK; // 2048
    tensor_dlt_kernel<<<blocks, BLK, 0, stream>>>(ca, h4, out, B);
}